// MambaBlock_9861244912039
// MI455X (gfx1250) — compile-verified
//
#include <hip/hip_runtime.h>
#include <math.h>

// ---------------- problem constants (from reference) ----------------
#define D_MODEL  1024
#define D_INNER  2048
#define D_STATE  16
#define D_CONV   4
#define DT_RANK  64
#define BATCH    2
#define SEQLEN   2048
#define NROWS    (BATCH * SEQLEN)         // 4096 token rows
#define XP_COLS  (DT_RANK + 2 * D_STATE)  // 96
#define LN_EPS   1e-5f

typedef __attribute__((ext_vector_type(2))) float v2f;
typedef __attribute__((ext_vector_type(8))) float v8f;

// ====================================================================
// LayerNorm: one block per token row (1024 cols, 256 threads, float4)
// ====================================================================
__global__ __launch_bounds__(256) void ln_kernel(const float* __restrict__ x,
                                                 const float* __restrict__ g,
                                                 const float* __restrict__ b,
                                                 float* __restrict__ xn) {
    const int row = blockIdx.x;
    const int tid = threadIdx.x;
    const float4 v  = ((const float4*)(x + (size_t)row * D_MODEL))[tid];
    float s  = v.x + v.y + v.z + v.w;
    float ss = v.x * v.x + v.y * v.y + v.z * v.z + v.w * v.w;
    #pragma unroll
    for (int off = 16; off > 0; off >>= 1) {
        s  += __shfl_xor(s,  off, 32);
        ss += __shfl_xor(ss, off, 32);
    }
    __shared__ float ls[8], lss[8];
    const int w = tid >> 5, lane = tid & 31;
    if (lane == 0) { ls[w] = s; lss[w] = ss; }
    __syncthreads();
    s = 0.f; ss = 0.f;
    #pragma unroll
    for (int i = 0; i < 8; i++) { s += ls[i]; ss += lss[i]; }
    const float mu  = s * (1.0f / D_MODEL);
    const float var = ss * (1.0f / D_MODEL) - mu * mu;
    const float inv = rsqrtf(var + LN_EPS);
    const float4 gv = ((const float4*)g)[tid];
    const float4 bv = ((const float4*)b)[tid];
    float4 o;
    o.x = (v.x - mu) * inv * gv.x + bv.x;
    o.y = (v.y - mu) * inv * gv.y + bv.y;
    o.z = (v.z - mu) * inv * gv.z + bv.z;
    o.w = (v.w - mu) * inv * gv.w + bv.w;
    ((float4*)(xn + (size_t)row * D_MODEL))[tid] = o;
}

// ====================================================================
// FP32 WMMA GEMM: C[M x N] = A[M x K] * B[K x N] (+ epilogue)
// Block = 256 threads (8 waves). C-tile = 128 x BN, wave strip 16 x BN.
// K staged through double-buffered LDS in chunks of 16
// (4 x V_WMMA_F32_16X16X4_F32 per chunk), B stored fragment-ready:
// pair-row p holds rows {2p, 2p+1} interleaved so each lane's B fragment
// is a single aligned ds_load_b64 into an even VGPR pair.
// EPI: 0 = store, 1 = softplus(acc + bias[n]), 2 = acc + res[m][n]
// ====================================================================
template <int NT, int EPI>
__global__ __launch_bounds__(256) void wmma_gemm(const float* __restrict__ A,
                                                 const float* __restrict__ B,
                                                 float* __restrict__ C,
                                                 const float* __restrict__ epi,
                                                 int K, int lda, int ldb, int ldc,
                                                 int ld_epi) {
    constexpr int BN  = NT * 16;
    constexpr int SA  = 20;            // padded A stride (conflict-free frag reads)
    constexpr int SBP = 2 * BN + 32;   // pair-row stride; SBP % 64 == 32
    constexpr int BVEC = ((16 * BN) % 1024 == 0);  // float4 B staging possible?
    constexpr int BCNT = BVEC ? (16 * BN) / 1024 : (16 * BN) / 256;

    __shared__ float Alds[2][128 * SA];
    __shared__ float Blds[2][8 * SBP];

    const int tid  = threadIdx.x;
    const int wave = tid >> 5;
    const int lane = tid & 31;
    const int lh   = lane >> 4;   // half-wave: K-pair of A / pair-row of B
    const int ll   = lane & 15;   // M (A) / N (B,C) index within tile
    const int mBase = blockIdx.y * 128;
    const int nBase = blockIdx.x * BN;

    v8f acc[NT];
    const v8f zero = {0.f, 0.f, 0.f, 0.f, 0.f, 0.f, 0.f, 0.f};
    #pragma unroll
    for (int i = 0; i < NT; i++) acc[i] = zero;

    // ---- staging registers for software pipeline ----
    float4 aReg[2];
    float4 bReg4[BVEC ? BCNT : 1];
    float  bRegS[BVEC ? 1 : BCNT];

    auto loadRegs = [&](int k0) {
        // A tile 128x16 as float4: 512 vectors, 2 per thread
        #pragma unroll
        for (int u = 0; u < 2; u++) {
            const int idx = u * 256 + tid;
            const int r = idx >> 2, c = (idx & 3) * 4;
            aReg[u] = *(const float4*)&A[(size_t)(mBase + r) * lda + (k0 + c)];
        }
        if constexpr (BVEC) {
            #pragma unroll
            for (int u = 0; u < BCNT; u++) {
                const int idx = u * 256 + tid;
                const int r = idx / (BN / 4), n = (idx % (BN / 4)) * 4;
                bReg4[u] = *(const float4*)&B[(size_t)(k0 + r) * ldb + (nBase + n)];
            }
        } else {
            #pragma unroll
            for (int u = 0; u < BCNT; u++) {
                const int i = u * 256 + tid;
                const int r = i / BN, n = i % BN;
                bRegS[u] = B[(size_t)(k0 + r) * ldb + (nBase + n)];
            }
        }
    };
    auto storeRegs = [&](int buf) {
        #pragma unroll
        for (int u = 0; u < 2; u++) {
            const int idx = u * 256 + tid;
            const int r = idx >> 2, c = (idx & 3) * 4;
            *(float4*)&Alds[buf][r * SA + c] = aReg[u];
        }
        if constexpr (BVEC) {
            #pragma unroll
            for (int u = 0; u < BCNT; u++) {
                const int idx = u * 256 + tid;
                const int r = idx / (BN / 4), n = (idx % (BN / 4)) * 4;
                const int p = r >> 1, kl = r & 1;
                Blds[buf][p * SBP + 2 * (n + 0) + kl] = bReg4[u].x;
                Blds[buf][p * SBP + 2 * (n + 1) + kl] = bReg4[u].y;
                Blds[buf][p * SBP + 2 * (n + 2) + kl] = bReg4[u].z;
                Blds[buf][p * SBP + 2 * (n + 3) + kl] = bReg4[u].w;
            }
        } else {
            #pragma unroll
            for (int u = 0; u < BCNT; u++) {
                const int i = u * 256 + tid;
                const int r = i / BN, n = i % BN;
                Blds[buf][(r >> 1) * SBP + 2 * n + (r & 1)] = bRegS[u];
            }
        }
    };

    // ---- prologue: fill buffer 0 ----
    loadRegs(0);
    storeRegs(0);
    int cur = 0;

    for (int k0 = 0; k0 < K; k0 += 16) {
        const bool last = (k0 + 16 >= K);
        if (!last) loadRegs(k0 + 16);        // overlap global latency w/ compute
        __syncthreads();                     // cur-buffer stores visible; prior
                                             // reads of other buffer drained
        #pragma unroll
        for (int kk = 0; kk < 4; kk++) {
            const int arow = wave * 16 + ll;
            const v2f a = *(const v2f*)&Alds[cur][arow * SA + kk * 4 + lh * 2];
            #pragma unroll
            for (int nt = 0; nt < NT; nt++) {
                const v2f bf =
                    *(const v2f*)&Blds[cur][(kk * 2 + lh) * SBP + 2 * (nt * 16 + ll)];
                acc[nt] = __builtin_amdgcn_wmma_f32_16x16x4_f32(
                    false, a, false, bf, (short)0, acc[nt], false, false);
            }
        }
        if (!last) { storeRegs(cur ^ 1); cur ^= 1; }
    }

    // Epilogue: D layout — component j -> M = j + 8*lh, N = ll (+tile bases)
    #pragma unroll
    for (int nt = 0; nt < NT; nt++) {
        #pragma unroll
        for (int j = 0; j < 8; j++) {
            const int m = mBase + wave * 16 + j + lh * 8;
            const int n = nBase + nt * 16 + ll;
            float v = acc[nt][j];
            if (EPI == 1) {                       // bias + softplus
                v += epi[n];
                v = (v > 20.f) ? v : __logf(1.f + __expf(v));
            } else if (EPI == 2) {                // + residual
                v += epi[(size_t)m * ld_epi + n];
            }
            C[(size_t)m * ldc + n] = v;
        }
    }
}

// ====================================================================
// Depthwise causal conv (4 taps) + bias + SiLU, reading xb half of xz
// ====================================================================
__global__ __launch_bounds__(256) void conv_silu_kernel(const float* __restrict__ xz,
                                                        const float* __restrict__ cw,
                                                        const float* __restrict__ cb,
                                                        float* __restrict__ xb) {
    const int d = blockIdx.x * 256 + threadIdx.x;  // channel
    const int t = blockIdx.y;                      // time
    const int b = blockIdx.z;                      // batch
    const float w0 = cw[d * 4 + 0], w1 = cw[d * 4 + 1],
                w2 = cw[d * 4 + 2], w3 = cw[d * 4 + 3];
    const float* base = xz + (size_t)b * SEQLEN * (2 * D_INNER) + d;
    float acc = cb[d];
    if (t >= 3) acc += w0 * base[(size_t)(t - 3) * (2 * D_INNER)];
    if (t >= 2) acc += w1 * base[(size_t)(t - 2) * (2 * D_INNER)];
    if (t >= 1) acc += w2 * base[(size_t)(t - 1) * (2 * D_INNER)];
    acc += w3 * base[(size_t)t * (2 * D_INNER)];
    const float s = acc / (1.f + __expf(-acc));    // SiLU
    xb[((size_t)(b * SEQLEN + t)) * D_INNER + d] = s;
}

// ====================================================================
// Selective scan, lane-parallel over D_STATE:
//   thread (group, lane n) owns state n of one channel d.
//   65536 threads = 2048 waves -> ~8 waves/SIMD to hide load latency.
//   Recurrence critical path per step = v_exp_f32 + FMA only; the
//   y-reduction (4x shfl_xor inside the 16-lane group) is off-path.
// ====================================================================
__global__ __launch_bounds__(256) void scan_kernel(const float* __restrict__ xb,
                                                   const float* __restrict__ dt,
                                                   const float* __restrict__ xp,
                                                   const float* __restrict__ A_log,
                                                   const float* __restrict__ state0,
                                                   float* __restrict__ ys,
                                                   float* __restrict__ fstate) {
    const int tid  = threadIdx.x;
    const int n    = tid & 15;                       // state index
    const int dgrp = tid >> 4;                       // channel group in block
    const int d    = blockIdx.x * 16 + dgrp;         // channel
    const int b    = blockIdx.y;                     // batch

    const float Ac = -__expf(A_log[(size_t)d * D_STATE + n]);
    float st = state0[((size_t)(b * D_INNER + d)) * D_STATE + n];

    const float* dtp = dt + (size_t)b * SEQLEN * D_INNER + d;
    const float* xbp = xb + (size_t)b * SEQLEN * D_INNER + d;
    const float* bcp = xp + (size_t)b * SEQLEN * XP_COLS;
    float*       ysp = ys + (size_t)b * SEQLEN * D_INNER + d;

    #pragma unroll 4
    for (int t = 0; t < SEQLEN; t++) {
        const float dtv = dtp[(size_t)t * D_INNER];          // group-uniform
        const float xv  = xbp[(size_t)t * D_INNER];          // group-uniform
        const float Bn  = bcp[(size_t)t * XP_COLS + DT_RANK + n];
        const float Cn  = bcp[(size_t)t * XP_COLS + DT_RANK + D_STATE + n];
        const float dA  = __expf(dtv * Ac);
        st = dA * st + (dtv * xv) * Bn;
        float prod = st * Cn;
        // reduce over the 16-lane state group (stays inside half-wave)
        #pragma unroll
        for (int off = 1; off < 16; off <<= 1)
            prod += __shfl_xor(prod, off, 32);
        if (n == 0) ysp[(size_t)t * D_INNER] = prod;
    }
    fstate[((size_t)(b * D_INNER + d)) * D_STATE + n] = st;
}

// ====================================================================
// Gate: y = (ys + xb*D_skip) * silu(z), in place over ys
// ====================================================================
__global__ __launch_bounds__(256) void gate_kernel(const float* __restrict__ xz,
                                                   const float* __restrict__ xb,
                                                   const float* __restrict__ Dsk,
                                                   float* __restrict__ ys) {
    const size_t idx = (size_t)blockIdx.x * 256 + threadIdx.x;
    const int d = (int)(idx & (D_INNER - 1));
    const size_t row = idx >> 11;                  // / D_INNER
    float v = ys[idx] + xb[idx] * Dsk[d];
    const float z = xz[row * (2 * D_INNER) + D_INNER + d];
    v *= z / (1.f + __expf(-z));
    ys[idx] = v;
}

// ====================================================================
extern "C" void kernel_launch(void* const* d_in, const int* in_sizes, int n_in,
                              void* d_out, int out_size, void* d_ws, size_t ws_size,
                              hipStream_t stream) {
    const float* x      = (const float*)d_in[0];
    const float* state  = (const float*)d_in[1];
    const float* ln_g   = (const float*)d_in[2];
    const float* ln_b   = (const float*)d_in[3];
    const float* W_in   = (const float*)d_in[4];
    const float* conv_w = (const float*)d_in[5];
    const float* conv_b = (const float*)d_in[6];
    const float* W_xp   = (const float*)d_in[7];
    const float* W_dt   = (const float*)d_in[8];
    const float* b_dt   = (const float*)d_in[9];
    const float* A_log  = (const float*)d_in[10];
    const float* D_skip = (const float*)d_in[11];
    const float* W_out  = (const float*)d_in[12];
    float* out = (float*)d_out;

    // workspace carve-up (floats)
    float* ws = (float*)d_ws;
    float* xn = ws;                                      //  4096*1024
    float* xz = xn + (size_t)NROWS * D_MODEL;            //  4096*4096
    float* xb = xz + (size_t)NROWS * 2 * D_INNER;        //  4096*2048
    float* xp = xb + (size_t)NROWS * D_INNER;            //  4096*96
    float* dt = xp + (size_t)NROWS * XP_COLS;            //  4096*2048
    float* ys = dt + (size_t)NROWS * D_INNER;            //  4096*2048
    float* fstate = out + (size_t)NROWS * D_MODEL;       // tail of d_out

    // 1) LayerNorm
    ln_kernel<<<NROWS, 256, 0, stream>>>(x, ln_g, ln_b, xn);

    // 2) xz = xn @ W_in   (4096x1024 * 1024x4096)
    wmma_gemm<8, 0><<<dim3(4096 / 128, NROWS / 128), 256, 0, stream>>>(
        xn, W_in, xz, nullptr, D_MODEL, D_MODEL, 2 * D_INNER, 2 * D_INNER, 0);

    // 3) depthwise conv + SiLU -> xb
    conv_silu_kernel<<<dim3(D_INNER / 256, SEQLEN, BATCH), 256, 0, stream>>>(
        xz, conv_w, conv_b, xb);

    // 4) xp = xb @ W_xp   (4096x2048 * 2048x96)
    wmma_gemm<6, 0><<<dim3(1, NROWS / 128), 256, 0, stream>>>(
        xb, W_xp, xp, nullptr, D_INNER, D_INNER, XP_COLS, XP_COLS, 0);

    // 5) dt = softplus(dt_lo @ W_dt + b_dt)   (4096x64 * 64x2048)
    wmma_gemm<8, 1><<<dim3(D_INNER / 128, NROWS / 128), 256, 0, stream>>>(
        xp, W_dt, dt, b_dt, DT_RANK, XP_COLS, D_INNER, D_INNER, 0);

    // 6) selective scan (lane-parallel states) -> ys, final state -> d_out tail
    scan_kernel<<<dim3(D_INNER / 16, BATCH), 256, 0, stream>>>(
        xb, dt, xp, A_log, state, ys, fstate);

    // 7) gate (in place over ys)
    gate_kernel<<<(NROWS * D_INNER) / 256, 256, 0, stream>>>(xz, xb, D_skip, ys);

    // 8) out = y @ W_out + residual   (4096x2048 * 2048x1024)
    wmma_gemm<8, 2><<<dim3(D_MODEL / 128, NROWS / 128), 256, 0, stream>>>(
        ys, W_out, out, x, D_INNER, D_INNER, D_MODEL, D_MODEL, D_MODEL);
}